// GcnRankerNet_3169685865284
// MI455X (gfx1250) — compile-verified
//
#include <hip/hip_runtime.h>
#include <math.h>

// ---------------- model constants ----------------
#define D_IN   768
#define D_H    128
#define G4     256   // 4 * H_LSTM gates
#define H_L    64    // hidden per direction

typedef __attribute__((ext_vector_type(2))) float v2f;
typedef __attribute__((ext_vector_type(8))) float v8f;

// ---------------- utility kernels ----------------
__global__ void k_zero(float* p, int n) {
    int i = blockIdx.x * blockDim.x + threadIdx.x;
    if (i < n) p[i] = 0.0f;
}

// deg[col[e]] += w[e]
__global__ void k_deg(const int* col, const float* w, float* deg, int nE) {
    int e = blockIdx.x * blockDim.x + threadIdx.x;
    if (e < nE) atomicAdd(&deg[col[e]], w[e]);
}

// in-place: deg -> deg>0 ? rsqrt(deg) : 0
__global__ void k_dis(float* deg, int n) {
    int i = blockIdx.x * blockDim.x + threadIdx.x;
    if (i < n) {
        float d = deg[i];
        deg[i] = (d > 0.0f) ? rsqrtf(d) : 0.0f;
    }
}

// ---------------- FP32 WMMA GEMM: C[M,N] = A[M,K] @ B[N,K]^T (+bias1+bias2)(+relu) ----
// One wave computes a 16 x (16*NT) strip via V_WMMA_F32_16X16X4_F32.
// A fragment (16x4): lanes 0-15 hold M=lane, K={k,k+1}; lanes 16-31 hold K={k+2,k+3}.
// B fragment (4x16): same striping indexed by N (B row-major [N,K], i.e. transposed).
// D fragment: VGPR r -> lanes 0-15: (M=m0+r, N=n0+lane); lanes 16-31: (M=m0+r+8).
// K processed in blocks of 32 (8 WMMAs): A double-buffered across blocks, B batched
// per block so the loadcnt wait amortizes over 8 back-to-back WMMAs.
template <int K, int NT, int ACT>
__global__ void k_wmma_gemm(const float* __restrict__ A, const float* __restrict__ B,
                            const float* __restrict__ bias1, const float* __restrict__ bias2,
                            float* __restrict__ C, int N) {
    const int lane = threadIdx.x;        // 0..31, all active -> EXEC == ~0
    const int half = lane >> 4;          // 0 or 1
    const int l16  = lane & 15;
    const int m0 = blockIdx.y * 16;
    const int n0 = blockIdx.x * (16 * NT);

    const float* Arow = A + (size_t)(m0 + l16) * K + half * 2;   // 8B aligned
    const float* Brow = B + (size_t)(n0 + l16) * K + half * 2;

    const v8f vzero = {};
    v8f acc[NT];
#pragma unroll
    for (int t = 0; t < NT; ++t) acc[t] = vzero;

    // prologue: A block 0
    v2f a[8];
#pragma unroll
    for (int i = 0; i < 8; ++i) a[i] = *(const v2f*)(Arow + i * 4);

    for (int k = 0; k < K; k += 32) {
        // double-buffer next A block (clamped redundant reload on last iter: branchless)
        const int kn = (k + 32 < K) ? (k + 32) : k;
        v2f an[8];
#pragma unroll
        for (int i = 0; i < 8; ++i) an[i] = *(const v2f*)(Arow + kn + i * 4);
        // speculative stream-ahead prefetch (dropped if past allocation)
        __builtin_prefetch(Arow + k + 256, 0, 1);     // global_prefetch_b8

#pragma unroll
        for (int t = 0; t < NT; ++t) {
            const float* Bt = Brow + (size_t)t * 16 * K + k;
            v2f b[8];
#pragma unroll
            for (int i = 0; i < 8; ++i) b[i] = *(const v2f*)(Bt + i * 4);
#pragma unroll
            for (int i = 0; i < 8; ++i)
                acc[t] = __builtin_amdgcn_wmma_f32_16x16x4_f32(
                    /*neg_a=*/false, a[i], /*neg_b=*/false, b[i],
                    /*c_mod=*/(short)0, acc[t], /*reuse_a=*/false, /*reuse_b=*/false);
        }
#pragma unroll
        for (int i = 0; i < 8; ++i) a[i] = an[i];
    }

#pragma unroll
    for (int t = 0; t < NT; ++t) {
        const int n = n0 + t * 16 + l16;
        float bv = 0.0f;
        if (bias1) bv += bias1[n];
        if (bias2) bv += bias2[n];
#pragma unroll
        for (int r = 0; r < 8; ++r) {
            int m = m0 + r + half * 8;
            float v = acc[t][r] + bv;
            if (ACT) v = fmaxf(v, 0.0f);
            C[(size_t)m * N + n] = v;
        }
    }
}

// ---------------- GCN message scatter ----------------
// agg[col[e], :] += dis[row[e]]*w[e]*dis[col[e]] * xl[row[e], :]
__global__ void k_scatter(const float* __restrict__ xl, const int* __restrict__ row,
                          const int* __restrict__ col, const float* __restrict__ w,
                          const float* __restrict__ dis, float* __restrict__ agg, int nE) {
    const int tid = threadIdx.x;                 // feature 0..127
    const int e0  = blockIdx.x * 8;
#pragma unroll
    for (int i = 0; i < 8; ++i) {
        int e = e0 + i;
        if (e >= nE) break;
        int r = row[e], c = col[e];              // uniform -> scalar loads
        float nrm = dis[r] * w[e] * dis[c];
        float v = nrm * xl[(size_t)r * D_H + tid];
        atomicAdd(&agg[(size_t)c * D_H + tid], v);
    }
}

// in-place: a[i] = relu(a[i] + b[i % 128])
__global__ void k_bias_relu(float* a, const float* b, int n) {
    int i = blockIdx.x * blockDim.x + threadIdx.x;
    if (i < n) a[i] = fmaxf(a[i] + b[i & (D_H - 1)], 0.0f);
}

// ---------------- bidirectional LSTM recurrence ----------------
// XW = Wih @ x_t + bih + bhh precomputed for all t. block 256, grid 2 (fwd/bwd).
__global__ void __launch_bounds__(256, 1)
k_lstm_bidir(const float* __restrict__ XWf, const float* __restrict__ XWb,
             const float* __restrict__ Whh_f, const float* __restrict__ Whh_b,
             float* __restrict__ hf, float* __restrict__ hb, int n) {
    const bool fwd = (blockIdx.x == 0);
    const float* XW  = fwd ? XWf  : XWb;
    const float* Whh = fwd ? Whh_f : Whh_b;
    float* hout      = fwd ? hf   : hb;
    const int tid = threadIdx.x;                 // gate row 0..255

    float wrow[H_L];
#pragma unroll
    for (int j = 0; j < H_L; ++j) wrow[j] = Whh[(size_t)tid * H_L + j];

    __shared__ float hsh[H_L];
    __shared__ float csh[H_L];
    __shared__ float gsh[G4];
    if (tid < H_L) { hsh[tid] = 0.0f; csh[tid] = 0.0f; }
    __syncthreads();

    for (int s = 0; s < n; ++s) {
        const int t = fwd ? s : (n - 1 - s);
        float g = XW[(size_t)t * G4 + tid];
#pragma unroll
        for (int j = 0; j < H_L; ++j) g = fmaf(wrow[j], hsh[j], g);
        gsh[tid] = g;
        __syncthreads();
        if (tid < H_L) {
            float ig = 1.0f / (1.0f + expf(-gsh[tid]));
            float fg = 1.0f / (1.0f + expf(-gsh[H_L + tid]));
            float gg = tanhf(gsh[2 * H_L + tid]);
            float og = 1.0f / (1.0f + expf(-gsh[3 * H_L + tid]));
            float c  = fg * csh[tid] + ig * gg;
            float h  = og * tanhf(c);
            csh[tid] = c; hsh[tid] = h;
            hout[(size_t)t * H_L + tid] = h;
        }
        __syncthreads();
    }
}

// ---------------- output head: sigmoid(hf.Wl[0:64] + hb.Wl[64:128] + bl) ----------
__global__ void k_head(const float* __restrict__ hf, const float* __restrict__ hb,
                       const float* __restrict__ Wl, const float* __restrict__ bl,
                       float* __restrict__ out, int n) {
    int t = blockIdx.x * blockDim.x + threadIdx.x;
    if (t >= n) return;
    float acc = bl[0];
#pragma unroll 8
    for (int j = 0; j < H_L; ++j) acc = fmaf(hf[(size_t)t * H_L + j], Wl[j], acc);
#pragma unroll 8
    for (int j = 0; j < H_L; ++j) acc = fmaf(hb[(size_t)t * H_L + j], Wl[H_L + j], acc);
    out[t] = 1.0f / (1.0f + expf(-acc));
}

// ---------------- launcher ----------------
extern "C" void kernel_launch(void* const* d_in, const int* in_sizes, int n_in,
                              void* d_out, int out_size, void* d_ws, size_t ws_size,
                              hipStream_t stream) {
    const float* x     = (const float*)d_in[0];
    const int*   ei    = (const int*)  d_in[1];
    const float* w     = (const float*)d_in[2];
    const float* W1    = (const float*)d_in[3];
    const float* b1    = (const float*)d_in[4];
    const float* W2    = (const float*)d_in[5];
    const float* b2    = (const float*)d_in[6];
    const float* Wih_f = (const float*)d_in[7];
    const float* Whh_f = (const float*)d_in[8];
    const float* bih_f = (const float*)d_in[9];
    const float* bhh_f = (const float*)d_in[10];
    const float* Wih_b = (const float*)d_in[11];
    const float* Whh_b = (const float*)d_in[12];
    const float* bih_b = (const float*)d_in[13];
    const float* bhh_b = (const float*)d_in[14];
    const float* Wl    = (const float*)d_in[15];
    const float* bl    = (const float*)d_in[16];
    float* out = (float*)d_out;

    const int nN = in_sizes[0] / D_IN;       // 16384 nodes
    const int nE = in_sizes[2];              // 1048576 edges
    const int* row = ei;
    const int* col = ei + nE;

    // workspace layout (floats)
    float* ws   = (float*)d_ws;
    float* buf0 = ws;                                    // [nN * 128]  xl scratch
    float* buf1 = buf0 + (size_t)nN * D_H;               // [nN * 128]  agg/h scratch
    float* XWf  = buf1 + (size_t)nN * D_H;               // [nN * 256]
    float* XWb  = XWf  + (size_t)nN * G4;                // [nN * 256]
    float* hf   = XWb  + (size_t)nN * G4;                // [nN * 64]
    float* hb   = hf   + (size_t)nN * H_L;               // [nN * 64]
    float* deg  = hb   + (size_t)nN * H_L;               // [nN]  (becomes dis in-place)

    const int nF = nN * D_H;                 // 2,097,152 features per layer

    // degree + symmetric norm
    k_zero<<<(nN + 255) / 256, 256, 0, stream>>>(deg, nN);
    k_deg <<<(nE + 255) / 256, 256, 0, stream>>>(col, w, deg, nE);
    k_dis <<<(nN + 255) / 256, 256, 0, stream>>>(deg, nN);

    // GCN layer 1: xl = x @ W1^T ; agg = scatter(norm * xl[row] -> col) ; relu(+b1)
    k_wmma_gemm<D_IN, 4, 0><<<dim3(D_H / 64, nN / 16), 32, 0, stream>>>(
        x, W1, nullptr, nullptr, buf0, D_H);
    k_zero<<<(nF + 255) / 256, 256, 0, stream>>>(buf1, nF);
    k_scatter<<<(nE + 7) / 8, D_H, 0, stream>>>(buf0, row, col, w, deg, buf1, nE);
    k_bias_relu<<<(nF + 255) / 256, 256, 0, stream>>>(buf1, b1, nF);

    // GCN layer 2
    k_wmma_gemm<D_H, 4, 0><<<dim3(D_H / 64, nN / 16), 32, 0, stream>>>(
        buf1, W2, nullptr, nullptr, buf0, D_H);
    k_zero<<<(nF + 255) / 256, 256, 0, stream>>>(buf1, nF);   // buf1 free after GEMM2 (stream order)
    k_scatter<<<(nE + 7) / 8, D_H, 0, stream>>>(buf0, row, col, w, deg, buf1, nE);
    k_bias_relu<<<(nF + 255) / 256, 256, 0, stream>>>(buf1, b2, nF);

    // LSTM input pre-GEMMs: XW = h2 @ Wih^T + (bih + bhh), both directions
    k_wmma_gemm<D_H, 4, 0><<<dim3(G4 / 64, nN / 16), 32, 0, stream>>>(
        buf1, Wih_f, bih_f, bhh_f, XWf, G4);
    k_wmma_gemm<D_H, 4, 0><<<dim3(G4 / 64, nN / 16), 32, 0, stream>>>(
        buf1, Wih_b, bih_b, bhh_b, XWb, G4);

    // sequential recurrence, fwd + bwd concurrently
    k_lstm_bidir<<<2, 256, 0, stream>>>(XWf, XWb, Whh_f, Whh_b, hf, hb, nN);

    // head
    k_head<<<(nN + 255) / 256, 256, 0, stream>>>(hf, hb, Wl, bl, out, nN);
}